// DealerGraphSAGE_32787780338275
// MI455X (gfx1250) — compile-verified
//
#include <hip/hip_runtime.h>

typedef __attribute__((ext_vector_type(2))) float v2f;
typedef __attribute__((ext_vector_type(8))) float v8f;

#define WMMA_F32_4(a, b, c) \
    __builtin_amdgcn_wmma_f32_16x16x4_f32(false, (a), false, (b), (short)0, (c), false, false)

__device__ __forceinline__ void atomAddF32(float* p, float v) {
    unsafeAtomicAdd(p, v);   // -> global_atomic_add_f32 (no CAS loop)
}

// ---------------------------------------------------------------- zero scratch
__global__ void zero_f32(float* __restrict__ p, long n) {
    long i = (long)blockIdx.x * blockDim.x + threadIdx.x;
    long stride = (long)gridDim.x * blockDim.x;
    for (; i < n; i += stride) p[i] = 0.0f;
}

// ------------------------------------------------- layer-1 scatter: x + degree
__global__ void scatter_x_deg(const float* __restrict__ x, const int* __restrict__ src,
                              const int* __restrict__ dst, float* __restrict__ agg1,
                              float* __restrict__ deg, int E) {
    int e = blockIdx.x * blockDim.x + threadIdx.x;
    if (e >= E) return;
    int s = src[e], d = dst[e];
    atomAddF32(&deg[d], 1.0f);
    const float* xr = x + (long)s * 9;
    float* ar = agg1 + (long)d * 16;
#pragma unroll
    for (int c = 0; c < 9; ++c) atomAddF32(&ar[c], xr[c]);
}

__global__ void recip_deg(float* __restrict__ deg, int n) {
    int i = blockIdx.x * blockDim.x + threadIdx.x;
    if (i < n) deg[i] = 1.0f / fmaxf(deg[i], 1.0f);
}

// ------------------------------------------------------------- layer 1 (WMMA)
// h = relu( [mean1 | x] @ [W1_l ; W1_r] + b1 ),  K = 32 (9 used + pad, 9 used + pad)
__global__ __launch_bounds__(256) void layer1_wmma(
    const float* __restrict__ agg1, const float* __restrict__ invdeg,
    const float* __restrict__ x, const float* __restrict__ W1l,
    const float* __restrict__ b1, const float* __restrict__ W1r,
    float* __restrict__ h, int ntiles) {
    __shared__ float Bt[64][36];      // TRANSPOSED: [col][k], stride 36 (=4 mod 8)
    __shared__ float As[8][16][36];   // per-wave 16 nodes x K=32

    int lane = threadIdx.x & 31;
    int wave = threadIdx.x >> 5;

    for (int i = threadIdx.x; i < 32 * 64; i += 256) {
        int k = i >> 6, c = i & 63;
        float v = 0.0f;
        if (k < 9)                    v = W1l[k * 64 + c];
        else if (k >= 16 && k < 25)   v = W1r[(k - 16) * 64 + c];
        Bt[c][k] = v;
    }

    int tile  = blockIdx.x * 8 + wave;
    bool active = tile < ntiles;
    int node0 = tile << 4;
    if (active) {
        for (int i = lane; i < 16 * 32; i += 32) {
            int m = i >> 5, k = i & 31;
            int node = node0 + m;
            float v = 0.0f;
            if (k < 9)                  v = agg1[(long)node * 16 + k] * invdeg[node];
            else if (k >= 16 && k < 25) v = x[(long)node * 9 + (k - 16)];
            As[wave][m][k] = v;
        }
    }
    __syncthreads();
    if (!active) return;

    int m    = lane & 15;
    int n    = lane & 15;
    int koff = (lane < 16) ? 0 : 2;      // kb+koff always even -> 8B-aligned b64 LDS loads
    v8f c0 = {}, c1 = {}, c2 = {}, c3 = {};
#pragma unroll
    for (int ks = 0; ks < 8; ++ks) {
        int kk = ks * 4 + koff;
        v2f a  = *(const v2f*)&As[wave][m][kk];
        v2f b0 = *(const v2f*)&Bt[ 0 + n][kk];
        v2f b1v = *(const v2f*)&Bt[16 + n][kk];
        v2f b2v = *(const v2f*)&Bt[32 + n][kk];
        v2f b3v = *(const v2f*)&Bt[48 + n][kk];
        c0 = WMMA_F32_4(a, b0,  c0);
        c1 = WMMA_F32_4(a, b1v, c1);
        c2 = WMMA_F32_4(a, b2v, c2);
        c3 = WMMA_F32_4(a, b3v, c3);
    }

    int mhi = (lane < 16) ? 0 : 8;
#pragma unroll
    for (int r = 0; r < 8; ++r) {
        int row = node0 + r + mhi;
        float* hp = h + (long)row * 64;
        hp[ 0 + n] = fmaxf(c0[r] + b1[ 0 + n], 0.0f);
        hp[16 + n] = fmaxf(c1[r] + b1[16 + n], 0.0f);
        hp[32 + n] = fmaxf(c2[r] + b1[32 + n], 0.0f);
        hp[48 + n] = fmaxf(c3[r] + b1[48 + n], 0.0f);
    }
}

// ------------------------------------------------------------- layer 2 (WMMA)
// [t | self] = h @ [W2_l | W2_r];  t -> scratch (to be scattered),
// out = self + b2 written directly (aggregated term added later).
__global__ __launch_bounds__(256) void layer2_wmma(
    const float* __restrict__ h, const float* __restrict__ W2l,
    const float* __restrict__ W2r, const float* __restrict__ b2,
    float* __restrict__ t, float* __restrict__ out, int ntiles) {
    __shared__ float Bt[64][68];      // TRANSPOSED: [col][k], stride 68 (=4 mod 8)
    __shared__ float As[8][16][68];   // per-wave 16 nodes x K=64

    int lane = threadIdx.x & 31;
    int wave = threadIdx.x >> 5;

    for (int i = threadIdx.x; i < 64 * 64; i += 256) {
        int k = i >> 6, c = i & 63;
        Bt[c][k] = (c < 32) ? W2l[k * 32 + c] : W2r[k * 32 + (c - 32)];
    }

    int tile  = blockIdx.x * 8 + wave;
    bool active = tile < ntiles;
    int node0 = tile << 4;
    if (active) {
        for (int i = lane; i < 256; i += 32) {      // 16 rows x 16 float4
            int m = i >> 4, q = (i & 15) * 4;
            const float4 hv = *(const float4*)(h + (long)(node0 + m) * 64 + q);
            As[wave][m][q + 0] = hv.x;
            As[wave][m][q + 1] = hv.y;
            As[wave][m][q + 2] = hv.z;
            As[wave][m][q + 3] = hv.w;
        }
    }
    __syncthreads();
    if (!active) return;

    int m    = lane & 15;
    int n    = lane & 15;
    int koff = (lane < 16) ? 0 : 2;
    v8f c0 = {}, c1 = {}, c2 = {}, c3 = {};
#pragma unroll
    for (int ks = 0; ks < 16; ++ks) {
        int kk = ks * 4 + koff;
        v2f a  = *(const v2f*)&As[wave][m][kk];
        v2f b0 = *(const v2f*)&Bt[ 0 + n][kk];
        v2f b1v = *(const v2f*)&Bt[16 + n][kk];
        v2f b2v = *(const v2f*)&Bt[32 + n][kk];
        v2f b3v = *(const v2f*)&Bt[48 + n][kk];
        c0 = WMMA_F32_4(a, b0,  c0);
        c1 = WMMA_F32_4(a, b1v, c1);
        c2 = WMMA_F32_4(a, b2v, c2);
        c3 = WMMA_F32_4(a, b3v, c3);
    }

    int mhi = (lane < 16) ? 0 : 8;
#pragma unroll
    for (int r = 0; r < 8; ++r) {
        int row = node0 + r + mhi;
        t[(long)row * 32 + n]        = c0[r];
        t[(long)row * 32 + 16 + n]   = c1[r];
        out[(long)row * 32 + n]      = c2[r] + b2[n];
        out[(long)row * 32 + 16 + n] = c3[r] + b2[16 + n];
    }
}

// --------------------------------------- layer-2 scatter of transformed feats
__global__ void scatter_t(const float* __restrict__ t, const int* __restrict__ src,
                          const int* __restrict__ dst, float* __restrict__ agg2, int E) {
    long tid = (long)blockIdx.x * blockDim.x + threadIdx.x;
    if (tid >= (long)E * 8) return;
    int e = (int)(tid >> 3);
    int q = (int)(tid & 7) * 4;
    int s = src[e], d = dst[e];
    const float4 tv = *(const float4*)(t + (long)s * 32 + q);
    float* ap = agg2 + (long)d * 32 + q;
    atomAddF32(ap + 0, tv.x);
    atomAddF32(ap + 1, tv.y);
    atomAddF32(ap + 2, tv.z);
    atomAddF32(ap + 3, tv.w);
}

__global__ void final_add(const float* __restrict__ agg2, const float* __restrict__ invdeg,
                          float* __restrict__ out, int n_nodes) {
    long i = (long)blockIdx.x * blockDim.x + threadIdx.x;
    if (i >= (long)n_nodes * 32) return;
    int node = (int)(i >> 5);
    out[i] += agg2[i] * invdeg[node];
}

extern "C" void kernel_launch(void* const* d_in, const int* in_sizes, int n_in,
                              void* d_out, int out_size, void* d_ws, size_t ws_size,
                              hipStream_t stream) {
    const float* x   = (const float*)d_in[0];
    const int*   ei  = (const int*)d_in[1];
    const float* W1l = (const float*)d_in[2];
    const float* b1  = (const float*)d_in[3];
    const float* W1r = (const float*)d_in[4];
    const float* W2l = (const float*)d_in[5];
    const float* b2  = (const float*)d_in[6];
    const float* W2r = (const float*)d_in[7];
    float* out = (float*)d_out;

    int N = in_sizes[0] / 9;
    int E = in_sizes[1] / 2;
    const int* src = ei;
    const int* dst = ei + E;

    // workspace layout (floats): deg[N] agg1[16N] h[64N] t[32N] agg2[32N] = 145N
    float* ws   = (float*)d_ws;
    float* deg  = ws;
    float* agg1 = deg + N;
    float* h    = agg1 + (long)N * 16;
    float* t    = h + (long)N * 64;
    float* agg2 = t + (long)N * 32;

    zero_f32<<<2048, 256, 0, stream>>>(deg, (long)N * 17);   // deg + agg1 contiguous
    zero_f32<<<2048, 256, 0, stream>>>(agg2, (long)N * 32);

    scatter_x_deg<<<(E + 255) / 256, 256, 0, stream>>>(x, src, dst, agg1, deg, E);
    recip_deg<<<(N + 255) / 256, 256, 0, stream>>>(deg, N);

    int ntiles = (N + 15) / 16;           // N = 100000 -> 6250 exact tiles
    int blocks = (ntiles + 7) / 8;        // 8 waves (tiles) per 256-thread block
    layer1_wmma<<<blocks, 256, 0, stream>>>(agg1, deg, x, W1l, b1, W1r, h, ntiles);
    layer2_wmma<<<blocks, 256, 0, stream>>>(h, W2l, W2r, b2, t, out, ntiles);

    long s2 = (long)E * 8;
    scatter_t<<<(int)((s2 + 255) / 256), 256, 0, stream>>>(t, src, dst, agg2, E);
    final_add<<<(int)(((long)N * 32 + 255) / 256), 256, 0, stream>>>(agg2, deg, out, N);
}